// TemperatureGNN_60842506715481
// MI455X (gfx1250) — compile-verified
//
#include <hip/hip_runtime.h>

typedef __attribute__((ext_vector_type(2))) float v2f;
typedef __attribute__((ext_vector_type(8))) float v8f;

#define CIN 128
#define HID 128

// ---------------------------------------------------------------------------
// D[M x 128] = A[M x 128] * W[128 x 128]  (optionally + bias, ReLU)
// One wave computes one 16x16 output tile via 32 chained V_WMMA_F32_16X16X4_F32.
// A-frag (16x4 f32): lane holds A[row][kh*2], A[row][kh*2+1]; row=lane&15, kh=lane>>4
// B-frag (4x16 f32): lane holds W[kh*2][col], W[kh*2+1][col];  col=lane&15
// C/D (16x16 f32, 8 VGPRs): elem r -> (M = 8*kh + r, N = col)
// ---------------------------------------------------------------------------
template <bool BIAS_RELU>
__global__ __launch_bounds__(128) void gemm128_wmma(
    const float* __restrict__ A, const float* __restrict__ W,
    const float* __restrict__ bias, float* __restrict__ D, int M)
{
    const int lane   = threadIdx.x & 31;
    const int wv     = threadIdx.x >> 5;
    const int mTile  = blockIdx.y * 4 + wv;
    const int mTiles = (M + 15) >> 4;
    if (mTile >= mTiles) return;              // wave-uniform: EXEC stays all-ones

    const int l15   = lane & 15;
    const int khalf = lane >> 4;              // 0 or 1
    const int row   = mTile * 16 + l15;
    const int col   = blockIdx.x * 16 + l15;

    const float2* __restrict__ ap = (const float2*)(A + (size_t)row * CIN);
    const float*  __restrict__ bp = W + col;

    v8f acc = {};
#pragma unroll 8
    for (int k = 0; k < CIN; k += 4) {
        float2 av = ap[(k >> 1) + khalf];     // A[row][k+2kh .. k+2kh+1]
        v2f a; a[0] = av.x; a[1] = av.y;
        v2f b;
        b[0] = bp[(size_t)(k + khalf * 2)     * HID];
        b[1] = bp[(size_t)(k + khalf * 2 + 1) * HID];
        acc = __builtin_amdgcn_wmma_f32_16x16x4_f32(
            /*neg_a=*/false, a, /*neg_b=*/false, b,
            /*c_mod=*/(short)0, acc, /*reuse_a=*/false, /*reuse_b=*/false);
    }

    const int mBase = mTile * 16 + khalf * 8;
    float bv = 0.0f;
    if (BIAS_RELU) bv = bias[col];
#pragma unroll
    for (int r = 0; r < 8; ++r) {
        float v = acc[r];
        if (BIAS_RELU) v = fmaxf(v + bv, 0.0f);
        D[(size_t)(mBase + r) * HID + col] = v;
    }
}

// deg[i] = 1 (self-loop contribution)
__global__ void deg_init_k(float* __restrict__ deg, int N) {
    int i = blockIdx.x * blockDim.x + threadIdx.x;
    if (i < N) deg[i] = 1.0f;
}

// deg[dst[e]] += 1 for every edge
__global__ void deg_count_k(const int* __restrict__ dst, float* __restrict__ deg, int E) {
    int e = blockIdx.x * blockDim.x + threadIdx.x;
    if (e < E) atomicAdd(&deg[dst[e]], 1.0f);
}

// dinv[i] = rsqrt(deg[i])   (deg >= 1 always, so no zero-guard needed)
__global__ void dinv_k(const float* __restrict__ deg, float* __restrict__ dinv, int N) {
    int i = blockIdx.x * blockDim.x + threadIdx.x;
    if (i < N) dinv[i] = rsqrtf(deg[i]);
}

// acc[i][c] = h[i][c] * dinv[i]^2   (self-loop message, also zero-initializes acc)
__global__ void selfloop_init_k(const float* __restrict__ h, const float* __restrict__ dinv,
                                float* __restrict__ acc, int total) {
    int t = blockIdx.x * blockDim.x + threadIdx.x;
    if (t >= total) return;
    float di = dinv[t >> 7];
    acc[t] = h[t] * di * di;
}

// One wave per edge: coalesced float4 row gather from h[src], 4 f32 atomics/lane into acc[dst]
__global__ __launch_bounds__(256) void edge_scatter_k(
    const int* __restrict__ src, const int* __restrict__ dst,
    const float* __restrict__ h, const float* __restrict__ dinv,
    float* __restrict__ acc, int E)
{
    int wave = (int)((blockIdx.x * (size_t)blockDim.x + threadIdx.x) >> 5);
    int lane = threadIdx.x & 31;
    if (wave >= E) return;
    int s = src[wave];
    int d = dst[wave];
    float nrm = dinv[s] * dinv[d];
    float4 hv = ((const float4*)(h + (size_t)s * HID))[lane];
    float* ap = acc + (size_t)d * HID + lane * 4;
    atomicAdd(ap + 0, hv.x * nrm);
    atomicAdd(ap + 1, hv.y * nrm);
    atomicAdd(ap + 2, hv.z * nrm);
    atomicAdd(ap + 3, hv.w * nrm);
}

// in-place: acc = relu(acc + bias[c])
__global__ void bias_relu_k(float* __restrict__ acc, const float* __restrict__ bias, int total) {
    int t = blockIdx.x * blockDim.x + threadIdx.x;
    if (t >= total) return;
    acc[t] = fmaxf(acc[t] + bias[t & (HID - 1)], 0.0f);
}

// out[i] = dot(h2[i], W2[:,0]) + b2   — one wave per node, shfl_xor wave32 reduction
__global__ __launch_bounds__(256) void matvec_out_k(
    const float* __restrict__ h, const float* __restrict__ W2,
    const float* __restrict__ b2, float* __restrict__ out, int N)
{
    int wave = (int)((blockIdx.x * (size_t)blockDim.x + threadIdx.x) >> 5);
    int lane = threadIdx.x & 31;
    if (wave >= N) return;
    float4 hv = ((const float4*)(h + (size_t)wave * HID))[lane];
    float4 wv = ((const float4*)W2)[lane];
    float s = hv.x * wv.x + hv.y * wv.y + hv.z * wv.z + hv.w * wv.w;
#pragma unroll
    for (int off = 16; off >= 1; off >>= 1) s += __shfl_xor(s, off, 32);
    if (lane == 0) out[wave] = s + b2[0];
}

extern "C" void kernel_launch(void* const* d_in, const int* in_sizes, int n_in,
                              void* d_out, int out_size, void* d_ws, size_t ws_size,
                              hipStream_t stream) {
    const float* x   = (const float*)d_in[0];
    const int*   ei  = (const int*)  d_in[1];
    const float* Wg  = (const float*)d_in[2];
    const float* bg  = (const float*)d_in[3];
    const float* W1  = (const float*)d_in[4];
    const float* b1  = (const float*)d_in[5];
    const float* W2  = (const float*)d_in[6];
    const float* b2  = (const float*)d_in[7];
    float*       out = (float*)d_out;

    const int N = in_sizes[0] / CIN;     // 50000
    const int E = in_sizes[1] / 2;       // 800000
    const int* src = ei;
    const int* dst = ei + E;
    const int total = N * HID;

    // workspace layout: buf0 = h_pre -> h2 ; buf1 = acc -> h1 ; deg ; dinv
    float* buf0 = (float*)d_ws;
    float* buf1 = buf0 + (size_t)N * HID;
    float* deg  = buf1 + (size_t)N * HID;
    float* dinv = deg + N;

    dim3 gGemm(HID / 16, (((N + 15) / 16) + 3) / 4);   // n-tiles x m-tile groups (4 waves/block)

    // 1) h_pre = x @ W_gcn
    gemm128_wmma<false><<<gGemm, 128, 0, stream>>>(x, Wg, nullptr, buf0, N);

    // 2) symmetric degree normalization
    deg_init_k <<<(N + 255) / 256, 256, 0, stream>>>(deg, N);
    deg_count_k<<<(E + 255) / 256, 256, 0, stream>>>(dst, deg, E);
    dinv_k     <<<(N + 255) / 256, 256, 0, stream>>>(deg, dinv, N);

    // 3) propagate: self-loop term then edge scatter-add
    selfloop_init_k<<<(total + 255) / 256, 256, 0, stream>>>(buf0, dinv, buf1, total);
    edge_scatter_k <<<(E + 7) / 8, 256, 0, stream>>>(src, dst, buf0, dinv, buf1, E);

    // 4) + b_gcn, ReLU (in place -> h1)
    bias_relu_k<<<(total + 255) / 256, 256, 0, stream>>>(buf1, bg, total);

    // 5) h2 = relu(h1 @ W1 + b1)
    gemm128_wmma<true><<<gGemm, 128, 0, stream>>>(buf1, W1, b1, buf0, N);

    // 6) out = h2 @ W2 + b2
    matvec_out_k<<<(N + 7) / 8, 256, 0, stream>>>(buf0, W2, b2, out, N);
}